// GCNEncoder_89300960019180
// MI455X (gfx1250) — compile-verified
//
#include <hip/hip_runtime.h>

typedef __attribute__((ext_vector_type(2))) float v2f;
typedef __attribute__((ext_vector_type(8))) float v8f;

// ---------------------------------------------------------------------------
// Degree / normalization
// ---------------------------------------------------------------------------
__global__ void k_deg_init(float* __restrict__ deg, int N) {
    int i = blockIdx.x * blockDim.x + threadIdx.x;
    if (i < N) deg[i] = 1.0f;  // self-loop
}

__global__ void k_deg_count(const int* __restrict__ dst, float* __restrict__ deg, int E) {
    int e = blockIdx.x * blockDim.x + threadIdx.x;
    if (e < E) atomicAdd(&deg[dst[e]], 1.0f);
}

__global__ void k_dinv(const float* __restrict__ deg, float* __restrict__ dinv, int N) {
    int i = blockIdx.x * blockDim.x + threadIdx.x;
    if (i < N) dinv[i] = rsqrtf(deg[i]);  // deg >= 1 always (self-loops)
}

// ---------------------------------------------------------------------------
// GEMM1: h1raw[N,64] = x[N,128] @ W1[128,64]   (WMMA f32 16x16x4)
// One wave -> 16 rows x 64 cols.
// W1 staged in LDS pair-interleaved: lds[(k>>1)*128 + 2*n + (k&1)] = W1[k][n],
// so each B operand {W[kk][n], W[kk+1][n]} is ONE aligned ds_load_b64.
// ---------------------------------------------------------------------------
__global__ void __launch_bounds__(256) k_gemm1(const float* __restrict__ x,
                                               const float* __restrict__ W1,
                                               float* __restrict__ h1raw,
                                               int rowTiles) {
    __shared__ float ldsW[128 * 64];
    for (int i = threadIdx.x; i < 128 * 64; i += 256) {
        const int k = i >> 6;          // W1 row (K index)
        const int n = i & 63;          // W1 col
        ldsW[(k >> 1) * 128 + (n << 1) + (k & 1)] = W1[i];
    }
    __syncthreads();

    const int lane = threadIdx.x & 31;
    const int w = blockIdx.x * 8 + (threadIdx.x >> 5);
    if (w >= rowTiles) return;                    // wave-uniform: EXEC stays all-1s

    const int row0 = w * 16;
    const int m    = lane & 15;                   // M (A) / N (B,C,D) index
    const int half = lane >> 4;                   // K-half select

    v8f acc0 = {}, acc1 = {}, acc2 = {}, acc3 = {};
    const float* xrow = x + (size_t)(row0 + m) * 128;
    __builtin_prefetch(xrow, 0, 0);               // global_prefetch_b8

    for (int k = 0; k < 128; k += 4) {
        const int kk = k + 2 * half;              // even
        const v2f a = *(const v2f*)(xrow + kk);   // A 16x4: K=kk,kk+1 per lane half
        __builtin_prefetch(xrow + kk + 32, 0, 0);

        const v2f* wp = (const v2f*)(ldsW + ((k >> 1) + half) * 128);
        const v2f b0 = wp[m +  0];
        const v2f b1 = wp[m + 16];
        const v2f b2 = wp[m + 32];
        const v2f b3 = wp[m + 48];

        acc0 = __builtin_amdgcn_wmma_f32_16x16x4_f32(false, a, false, b0, (short)0, acc0, false, false);
        acc1 = __builtin_amdgcn_wmma_f32_16x16x4_f32(false, a, false, b1, (short)0, acc1, false, false);
        acc2 = __builtin_amdgcn_wmma_f32_16x16x4_f32(false, a, false, b2, (short)0, acc2, false, false);
        acc3 = __builtin_amdgcn_wmma_f32_16x16x4_f32(false, a, false, b3, (short)0, acc3, false, false);
    }

    float* out = h1raw + (size_t)row0 * 64;
    #pragma unroll
    for (int r = 0; r < 8; ++r) {
        const int row = r + 8 * half;             // C/D: VGPR r holds M=r / M=r+8
        out[row * 64 + m +  0] = acc0[r];
        out[row * 64 + m + 16] = acc1[r];
        out[row * 64 + m + 32] = acc2[r];
        out[row * 64 + m + 48] = acc3[r];
    }
}

// ---------------------------------------------------------------------------
// GEMM2: h2raw[N,32] = h1[N,64] @ W2[64,32]  (same scheme, C=32)
// ---------------------------------------------------------------------------
__global__ void __launch_bounds__(256) k_gemm2(const float* __restrict__ h1,
                                               const float* __restrict__ W2,
                                               float* __restrict__ h2raw,
                                               int rowTiles) {
    __shared__ float ldsW[64 * 32];
    for (int i = threadIdx.x; i < 64 * 32; i += 256) {
        const int k = i >> 5;
        const int n = i & 31;
        ldsW[(k >> 1) * 64 + (n << 1) + (k & 1)] = W2[i];
    }
    __syncthreads();

    const int lane = threadIdx.x & 31;
    const int w = blockIdx.x * 8 + (threadIdx.x >> 5);
    if (w >= rowTiles) return;

    const int row0 = w * 16;
    const int m    = lane & 15;
    const int half = lane >> 4;

    v8f acc0 = {}, acc1 = {};
    const float* hrow = h1 + (size_t)(row0 + m) * 64;
    __builtin_prefetch(hrow, 0, 0);

    for (int k = 0; k < 64; k += 4) {
        const int kk = k + 2 * half;
        const v2f a = *(const v2f*)(hrow + kk);

        const v2f* wp = (const v2f*)(ldsW + ((k >> 1) + half) * 64);
        const v2f b0 = wp[m +  0];
        const v2f b1 = wp[m + 16];

        acc0 = __builtin_amdgcn_wmma_f32_16x16x4_f32(false, a, false, b0, (short)0, acc0, false, false);
        acc1 = __builtin_amdgcn_wmma_f32_16x16x4_f32(false, a, false, b1, (short)0, acc1, false, false);
    }

    float* out = h2raw + (size_t)row0 * 32;
    #pragma unroll
    for (int r = 0; r < 8; ++r) {
        const int row = r + 8 * half;
        out[row * 32 + m +  0] = acc0[r];
        out[row * 32 + m + 16] = acc1[r];
    }
}

// ---------------------------------------------------------------------------
// Self-loop init: agg = dinv[i]^2 * hraw   (D = 1<<lgD features per node)
// ---------------------------------------------------------------------------
__global__ void k_selfloop(const float* __restrict__ hraw, const float* __restrict__ dinv,
                           float* __restrict__ agg, int total, int lgD) {
    int t = blockIdx.x * blockDim.x + threadIdx.x;
    if (t >= total) return;
    const int i = t >> lgD;
    const float di = dinv[i];
    agg[t] = hraw[t] * di * di;
}

// ---------------------------------------------------------------------------
// Edge scatter, 64 features: one wave per edge, float2 per lane
// ---------------------------------------------------------------------------
__global__ void k_scatter64(const int* __restrict__ src, const int* __restrict__ dst,
                            const float* __restrict__ dinv, const float* __restrict__ h,
                            float* __restrict__ agg, int E) {
    int t = blockIdx.x * blockDim.x + threadIdx.x;
    int e = t >> 5;
    if (e >= E) return;
    const int p = (t & 31) * 2;
    const int s = src[e], d = dst[e];
    const float nrm = dinv[s] * dinv[d];
    const float2 v = *(const float2*)(h + (size_t)s * 64 + p);
    float* ap = agg + (size_t)d * 64 + p;
    atomicAdd(ap + 0, v.x * nrm);
    atomicAdd(ap + 1, v.y * nrm);
}

// ---------------------------------------------------------------------------
// Edge scatter, 32 features: one wave per edge, one float per lane
// ---------------------------------------------------------------------------
__global__ void k_scatter32(const int* __restrict__ src, const int* __restrict__ dst,
                            const float* __restrict__ dinv, const float* __restrict__ h,
                            float* __restrict__ agg, int E) {
    int t = blockIdx.x * blockDim.x + threadIdx.x;
    int e = t >> 5;
    if (e >= E) return;
    const int f = t & 31;
    const int s = src[e], d = dst[e];
    const float nrm = dinv[s] * dinv[d];
    atomicAdd(agg + (size_t)d * 32 + f, h[(size_t)s * 32 + f] * nrm);
}

// ---------------------------------------------------------------------------
// Bias (+ReLU) epilogues
// ---------------------------------------------------------------------------
__global__ void k_bias_relu(float* __restrict__ a, const float* __restrict__ b, int total, int mask) {
    int t = blockIdx.x * blockDim.x + threadIdx.x;
    if (t >= total) return;
    const float v = a[t] + b[t & mask];
    a[t] = v > 0.0f ? v : 0.0f;
}

__global__ void k_bias_out(const float* __restrict__ a, const float* __restrict__ b,
                           float* __restrict__ out, int total, int mask) {
    int t = blockIdx.x * blockDim.x + threadIdx.x;
    if (t >= total) return;
    out[t] = a[t] + b[t & mask];
}

// ---------------------------------------------------------------------------
extern "C" void kernel_launch(void* const* d_in, const int* in_sizes, int n_in,
                              void* d_out, int out_size, void* d_ws, size_t ws_size,
                              hipStream_t stream) {
    const float* x    = (const float*)d_in[0];
    const int*   edge = (const int*)d_in[1];
    const float* W1   = (const float*)d_in[2];
    const float* b1   = (const float*)d_in[3];
    const float* W2   = (const float*)d_in[4];
    const float* b2   = (const float*)d_in[5];
    float* out = (float*)d_out;

    const int N = in_sizes[0] / 128;   // 100000
    const int E = in_sizes[1] / 2;     // 1280000
    const int* esrc = edge;            // edge_index[0]
    const int* edst = edge + E;        // edge_index[1]

    // Workspace layout (floats)
    float* ws     = (float*)d_ws;
    float* deg    = ws;                                 // [N]
    float* dinv   = ws + (size_t)N;                     // [N]
    float* h1raw  = ws + 2 * (size_t)N;                 // [N*64]
    float* agg1   = h1raw + (size_t)N * 64;             // [N*64] -> becomes h1 after relu
    float* h2raw  = agg1  + (size_t)N * 64;             // [N*32]
    float* agg2   = h1raw;                              // reuse (h1raw dead after layer-1 scatter)

    const int B = 256;
    const int rowTiles = N / 16;                        // 6250 (exact)

    // 1) normalization
    k_deg_init <<<(N + B - 1) / B, B, 0, stream>>>(deg, N);
    k_deg_count<<<(E + B - 1) / B, B, 0, stream>>>(edst, deg, E);
    k_dinv     <<<(N + B - 1) / B, B, 0, stream>>>(deg, dinv, N);

    // 2) layer 1: GEMM (WMMA f32) -> self-loop init -> edge scatter -> bias+relu
    k_gemm1<<<(rowTiles + 7) / 8, B, 0, stream>>>(x, W1, h1raw, rowTiles);
    {
        const int tot = N * 64;
        k_selfloop<<<(tot + B - 1) / B, B, 0, stream>>>(h1raw, dinv, agg1, tot, 6);
        const int st = E * 32;
        k_scatter64<<<(st + B - 1) / B, B, 0, stream>>>(esrc, edst, dinv, h1raw, agg1, E);
        k_bias_relu<<<(tot + B - 1) / B, B, 0, stream>>>(agg1, b1, tot, 63);
    }

    // 3) layer 2
    k_gemm2<<<(rowTiles + 7) / 8, B, 0, stream>>>(agg1, W2, h2raw, rowTiles);
    {
        const int tot = N * 32;
        k_selfloop<<<(tot + B - 1) / B, B, 0, stream>>>(h2raw, dinv, agg2, tot, 5);
        const int st = E * 32;
        k_scatter32<<<(st + B - 1) / B, B, 0, stream>>>(esrc, edst, dinv, h2raw, agg2, E);
        k_bias_out <<<(tot + B - 1) / B, B, 0, stream>>>(agg2, b2, out, tot, 31);
    }
}